// SS2D_74423193305317
// MI455X (gfx1250) — compile-verified
//
#include <hip/hip_runtime.h>
#include <math.h>

// ---------------- problem dims ----------------
#define CD   192          // channels
#define NS   16           // state dim
#define KD   4            // directions
#define DR   12           // dt rank
#define BS   2
#define HS   64
#define WS   64
#define LS   (HS*WS)      // 4096
#define MROWS (BS*LS)     // 8192 rows for all GEMMs
#define TWO_C (2*CD)      // 384
#define DPROJ 48          // 44 padded to 48 (row stride of proj buffer)
#define CH   64           // scan chunk length
#define NCH  (LS/CH)      // 64 chunks

typedef float v2f __attribute__((ext_vector_type(2)));
typedef float v8f __attribute__((ext_vector_type(8)));

// ---------------- WMMA f32 16x16x4 fragment helpers (wave32) ----------------
// A (16x4, MxK): lane holds M = lane&15 ; VGPR j holds K = kb + 2*(lane>=16) + j
__device__ __forceinline__ v2f lda_f32(const float* __restrict__ A, int lda,
                                       int m0, int kb, int lane) {
  int m = m0 + (lane & 15);
  int k = kb + ((lane >> 4) << 1);
  const float* p = A + (size_t)m * lda + k;
  v2f a; a.x = p[0]; a.y = p[1]; return a;
}
// B (4x16, KxN): lane holds N = lane&15 ; VGPR j holds K = kb + 2*(lane>=16) + j
__device__ __forceinline__ v2f ldb_f32(const float* __restrict__ Bm, int ldb,
                                       int kb, int n0, int lane) {
  int n = n0 + (lane & 15);
  int k = kb + ((lane >> 4) << 1);
  v2f b; b.x = Bm[(size_t)k * ldb + n]; b.y = Bm[(size_t)(k + 1) * ldb + n]; return b;
}
// B with column guard (for the 44-wide x_proj weight, padded to 48)
__device__ __forceinline__ v2f ldb_f32_guard(const float* __restrict__ Bm, int ldb,
                                             int kb, int n0, int lane, int nmax) {
  int n = n0 + (lane & 15);
  int k = kb + ((lane >> 4) << 1);
  v2f b;
  if (n < nmax) { b.x = Bm[(size_t)k * ldb + n]; b.y = Bm[(size_t)(k + 1) * ldb + n]; }
  else          { b.x = 0.0f; b.y = 0.0f; }
  return b;
}
// D/C (16x16 f32): VGPR r holds M = m0 + r + 8*(lane>=16), N = n0 + (lane&15)
__device__ __forceinline__ void st_d(float* __restrict__ D, int ldd,
                                     int m0, int n0, int lane, const v8f& c) {
  int n  = n0 + (lane & 15);
  int mh = m0 + ((lane >> 4) << 3);
#pragma unroll
  for (int r = 0; r < 8; ++r) D[(size_t)(mh + r) * ldd + n] = c[r];
}

__device__ __forceinline__ float silu_f(float x) { return x / (1.0f + __expf(-x)); }

// ---------------- K1: xz = x @ in_proj_w ; split into x_ssm / z ----------------
// One wave -> one 16-row M tile x four 16-col N tiles (A fragment reused 4x).
__global__ __launch_bounds__(32)
void k_inproj(const float* __restrict__ x, const float* __restrict__ w,
              float* __restrict__ xssm, float* __restrict__ zbuf) {
  int lane = threadIdx.x;
  int n0 = blockIdx.x * 64;      // 6 groups of 64 over 2C=384 (never straddles C split)
  int m0 = blockIdx.y * 16;
  v8f acc[4] = {};
  for (int kb = 0; kb < CD; kb += 4) {
    v2f a = lda_f32(x, CD, m0, kb, lane);
#pragma unroll
    for (int t = 0; t < 4; ++t) {
      v2f b = ldb_f32(w, TWO_C, kb, n0 + t * 16, lane);
      acc[t] = __builtin_amdgcn_wmma_f32_16x16x4_f32(false, a, false, b,
                                                     (short)0, acc[t], false, false);
    }
  }
#pragma unroll
  for (int t = 0; t < 4; ++t) {
    int nt = n0 + t * 16;
    if (nt < CD) st_d(xssm, CD, m0, nt, lane, acc[t]);
    else         st_d(zbuf, CD, m0, nt - CD, lane, acc[t]);
  }
}

// ---------------- K2: depthwise 3x3 conv + SiLU -> 4 direction copies ----------------
// xs layout: ((k*BS + b)*LS + l)*CD + c, scan order per direction.
__global__ void k_conv_dirs(const float* __restrict__ xssm, const float* __restrict__ cw,
                            const float* __restrict__ cb, float* __restrict__ xs) {
  int idx = blockIdx.x * blockDim.x + threadIdx.x;
  if (idx >= BS * LS * CD) return;
  int c = idx % CD;
  int s = (idx / CD) % LS;
  int b = idx / (CD * LS);
  int h = s >> 6, w = s & 63;
  float acc = cb[c];
#pragma unroll
  for (int dh = 0; dh < 3; ++dh) {
    int hh = h + dh - 1;
    if ((unsigned)hh >= (unsigned)HS) continue;
#pragma unroll
    for (int dw = 0; dw < 3; ++dw) {
      int ww = w + dw - 1;
      if ((unsigned)ww >= (unsigned)WS) continue;
      acc += xssm[(((size_t)b * HS + hh) * WS + ww) * CD + c] * cw[c * 9 + dh * 3 + dw];
    }
  }
  acc = silu_f(acc);
  int l2 = ((s & 63) << 6) | (s >> 6);   // H==W==64: transpose index (involution)
  xs[(((size_t)0 * BS + b) * LS + s)            * CD + c] = acc;
  xs[(((size_t)1 * BS + b) * LS + (LS - 1 - s)) * CD + c] = acc;
  xs[(((size_t)2 * BS + b) * LS + l2)           * CD + c] = acc;
  xs[(((size_t)3 * BS + b) * LS + (LS - 1 - l2))* CD + c] = acc;
}

// ---------------- K3: proj[k] = xs[k] @ x_proj_w[k]  (192 -> 44, padded 48) ----------------
// One wave handles all 3 padded N tiles for its M tile.
__global__ __launch_bounds__(32)
void k_proj(const float* __restrict__ xs, const float* __restrict__ xpw,
            float* __restrict__ proj) {
  int lane = threadIdx.x;
  int m0 = blockIdx.x * 16;
  int k  = blockIdx.y;
  const float* A  = xs  + (size_t)k * MROWS * CD;
  const float* Bw = xpw + (size_t)k * CD * 44;
  float*       D  = proj + (size_t)k * MROWS * DPROJ;
  v8f acc[3] = {};
  for (int kb = 0; kb < CD; kb += 4) {
    v2f a = lda_f32(A, CD, m0, kb, lane);
    v2f b0 = ldb_f32(Bw, 44, kb, 0, lane);
    v2f b1 = ldb_f32(Bw, 44, kb, 16, lane);
    v2f b2 = ldb_f32_guard(Bw, 44, kb, 32, lane, 44);
    acc[0] = __builtin_amdgcn_wmma_f32_16x16x4_f32(false, a, false, b0, (short)0, acc[0], false, false);
    acc[1] = __builtin_amdgcn_wmma_f32_16x16x4_f32(false, a, false, b1, (short)0, acc[1], false, false);
    acc[2] = __builtin_amdgcn_wmma_f32_16x16x4_f32(false, a, false, b2, (short)0, acc[2], false, false);
  }
#pragma unroll
  for (int t = 0; t < 3; ++t) st_d(D, DPROJ, m0, t * 16, lane, acc[t]);
}

// ---------------- K4: delta = softplus(proj[:, :12] @ dt_w + 2*dt_b) ----------------
// (reference adds dt_b once before and once inside softplus)
__global__ __launch_bounds__(32)
void k_delta(const float* __restrict__ proj, const float* __restrict__ dtw,
             const float* __restrict__ dtb, float* __restrict__ delta) {
  int lane = threadIdx.x;
  int n0 = blockIdx.x * 64;      // 3 groups of 64 over C=192
  int m0 = blockIdx.y * 16;
  int k  = blockIdx.z;
  const float* A  = proj + (size_t)k * MROWS * DPROJ;   // uses cols 0..11
  const float* Bw = dtw  + (size_t)k * DR * CD;
  float*       D  = delta + (size_t)k * MROWS * CD;
  v8f acc[4] = {};
  for (int kb = 0; kb < DR; kb += 4) {
    v2f a = lda_f32(A, DPROJ, m0, kb, lane);
#pragma unroll
    for (int t = 0; t < 4; ++t) {
      v2f b = ldb_f32(Bw, CD, kb, n0 + t * 16, lane);
      acc[t] = __builtin_amdgcn_wmma_f32_16x16x4_f32(false, a, false, b,
                                                     (short)0, acc[t], false, false);
    }
  }
  int mh = m0 + ((lane >> 4) << 3);
#pragma unroll
  for (int t = 0; t < 4; ++t) {
    int n = n0 + t * 16 + (lane & 15);
    float bias2 = 2.0f * dtb[k * CD + n];
#pragma unroll
    for (int r = 0; r < 8; ++r) {
      float v = acc[t][r] + bias2;
      v = (v > 20.0f) ? v : log1pf(__expf(v));        // softplus
      D[(size_t)(mh + r) * CD + n] = v;
    }
  }
}

// ---------------- K5a: chunked local scan (zero initial state) ----------------
// The diagonal linear recurrence s <- exp(d*An)*s + (d*u)*Bn is associative:
// over a chunk, s_end = exp(An*sum(d))*s_start + local. Run 64-step chunks in
// parallel (512 blocks instead of 8), write uncorrected y, the running sum of
// d (in place over delta), and the per-chunk final local state.
__global__ __launch_bounds__(CD)
void k_scan_local(const float* __restrict__ xs, const float* __restrict__ proj,
                  float* __restrict__ delta, const float* __restrict__ alog,
                  const float* __restrict__ dsv, float* __restrict__ ys,
                  float* __restrict__ chstate) {
  int c     = threadIdx.x;               // 0..191
  int chunk = blockIdx.x % NCH;
  int kb    = blockIdx.x / NCH;          // k*BS + b
  int k     = kb / BS;
  float An[NS];
#pragma unroll
  for (int n = 0; n < NS; ++n)
    An[n] = -__expf(alog[((size_t)k * CD + c) * NS + n]);   // A_neg
  float dscale = dsv[k * CD + c];
  float st[NS];
#pragma unroll
  for (int n = 0; n < NS; ++n) st[n] = 0.0f;

  size_t l0 = (size_t)kb * LS + (size_t)chunk * CH;
  const float* xrow = xs    + l0 * CD;
  float*       drow = delta + l0 * CD;
  const float* prow = proj  + l0 * DPROJ;
  float*       yrow = ys    + l0 * CD;

  float cum = 0.0f;
  for (int i = 0; i < CH; ++i) {
    if (i + 1 < CH) __builtin_prefetch(prow + (size_t)(i + 1) * DPROJ, 0, 0);
    float d = drow[(size_t)i * CD + c];
    cum += d;
    drow[(size_t)i * CD + c] = cum;                 // delta becomes cumd in place
    float u = xrow[(size_t)i * CD + c];
    const float* pr = prow + (size_t)i * DPROJ;
    float Bv[NS], Cv[NS];
#pragma unroll
    for (int n = 0; n < NS; ++n) { Bv[n] = pr[DR + n]; Cv[n] = pr[DR + NS + n]; }
    float du = d * u;
    float y = 0.0f;
#pragma unroll
    for (int n = 0; n < NS; ++n) {
      st[n] = __expf(d * An[n]) * st[n] + du * Bv[n];
      y += st[n] * Cv[n];
    }
    yrow[(size_t)i * CD + c] = y + dscale * u;
  }
  float* fs = chstate + (((size_t)kb * NCH + chunk) * CD + c) * NS;
#pragma unroll
  for (int n = 0; n < NS; ++n) fs[n] = st[n];
}

// ---------------- K5b: compose chunk carries (tiny sequential pass) ----------------
// s_start[0] = 0 ; s_start[j] = exp(An*sumd[j-1]) * s_start[j-1] + final[j-1]
__global__ __launch_bounds__(CD)
void k_scan_carry(const float* __restrict__ delta /*=cumd*/,
                  const float* __restrict__ chstate, const float* __restrict__ alog,
                  float* __restrict__ sstart) {
  int c  = threadIdx.x;
  int kb = blockIdx.x;                   // 0..7
  int k  = kb / BS;
  float An[NS];
#pragma unroll
  for (int n = 0; n < NS; ++n)
    An[n] = -__expf(alog[((size_t)k * CD + c) * NS + n]);
  float s0[NS];
#pragma unroll
  for (int n = 0; n < NS; ++n) s0[n] = 0.0f;
  {
    float* w0 = sstart + (((size_t)kb * NCH + 0) * CD + c) * NS;
#pragma unroll
    for (int n = 0; n < NS; ++n) w0[n] = 0.0f;
  }
  for (int j = 0; j < NCH - 1; ++j) {
    float sumd = delta[((size_t)kb * LS + (size_t)j * CH + (CH - 1)) * CD + c];
    const float* fj = chstate + (((size_t)kb * NCH + j) * CD + c) * NS;
    float* wj = sstart + (((size_t)kb * NCH + j + 1) * CD + c) * NS;
#pragma unroll
    for (int n = 0; n < NS; ++n) {
      s0[n] = __expf(sumd * An[n]) * s0[n] + fj[n];
      wj[n] = s0[n];
    }
  }
}

// ---------------- K5c: apply carry correction in parallel ----------------
// y_l += sum_n C[l,n] * exp(cumd[l]*An[n]) * s_start[n]
__global__ __launch_bounds__(CD)
void k_scan_fix(const float* __restrict__ proj, const float* __restrict__ delta /*=cumd*/,
                const float* __restrict__ alog, const float* __restrict__ sstart,
                float* __restrict__ ys) {
  int c     = threadIdx.x;
  int chunk = blockIdx.x % NCH;
  if (chunk == 0) return;                // s_start is zero for chunk 0
  int kb    = blockIdx.x / NCH;
  int k     = kb / BS;
  float An[NS];
#pragma unroll
  for (int n = 0; n < NS; ++n)
    An[n] = -__expf(alog[((size_t)k * CD + c) * NS + n]);
  float s0[NS];
  const float* sp = sstart + (((size_t)kb * NCH + chunk) * CD + c) * NS;
#pragma unroll
  for (int n = 0; n < NS; ++n) s0[n] = sp[n];

  size_t l0 = (size_t)kb * LS + (size_t)chunk * CH;
  const float* drow = delta + l0 * CD;
  const float* prow = proj  + l0 * DPROJ;
  float*       yrow = ys    + l0 * CD;
  for (int i = 0; i < CH; ++i) {
    float cd = drow[(size_t)i * CD + c];
    const float* pr = prow + (size_t)i * DPROJ;
    float corr = 0.0f;
#pragma unroll
    for (int n = 0; n < NS; ++n)
      corr += pr[DR + NS + n] * (__expf(cd * An[n]) * s0[n]);
    yrow[(size_t)i * CD + c] += corr;
  }
}

// ---------------- K6a: merge 4 directions + LayerNorm + SiLU(z) gate ----------------
__global__ __launch_bounds__(CD)
void k_merge_ln(const float* __restrict__ ys, const float* __restrict__ zbuf,
                const float* __restrict__ lng, const float* __restrict__ lnb,
                float* __restrict__ g) {
  int c  = threadIdx.x;        // 0..191
  int bs = blockIdx.x;         // b*LS + s (spatial row-major)
  int b  = bs / LS, s = bs % LS;
  int l2 = ((s & 63) << 6) | (s >> 6);
  float y =
      ys[(((size_t)0 * BS + b) * LS + s)             * CD + c]
    + ys[(((size_t)1 * BS + b) * LS + (LS - 1 - s))  * CD + c]
    + ys[(((size_t)2 * BS + b) * LS + l2)            * CD + c]
    + ys[(((size_t)3 * BS + b) * LS + (LS - 1 - l2)) * CD + c];

  // block reduction over 192 channels (6 wave32s)
  float v = y, v2 = y * y;
#pragma unroll
  for (int off = 16; off; off >>= 1) {
    v  += __shfl_xor(v,  off);
    v2 += __shfl_xor(v2, off);
  }
  __shared__ float s1[6], s2[6];
  __shared__ float mu_s, rs_s;
  int wid = c >> 5;
  if ((c & 31) == 0) { s1[wid] = v; s2[wid] = v2; }
  __syncthreads();
  if (c == 0) {
    float a = 0.0f, bb = 0.0f;
    for (int i = 0; i < 6; ++i) { a += s1[i]; bb += s2[i]; }
    float mu  = a / (float)CD;
    float var = bb / (float)CD - mu * mu;
    mu_s = mu; rs_s = rsqrtf(var + 1e-5f);
  }
  __syncthreads();
  float zn = zbuf[(size_t)bs * CD + c];
  g[(size_t)bs * CD + c] = ((y - mu_s) * rs_s * lng[c] + lnb[c]) * silu_f(zn);
}

// ---------------- K6b: out = g @ out_w ----------------
__global__ __launch_bounds__(32)
void k_out(const float* __restrict__ g, const float* __restrict__ ow,
           float* __restrict__ out) {
  int lane = threadIdx.x;
  int n0 = blockIdx.x * 64;      // 3 groups of 64 over C=192
  int m0 = blockIdx.y * 16;
  v8f acc[4] = {};
  for (int kb = 0; kb < CD; kb += 4) {
    v2f a = lda_f32(g, CD, m0, kb, lane);
#pragma unroll
    for (int t = 0; t < 4; ++t) {
      v2f b = ldb_f32(ow, CD, kb, n0 + t * 16, lane);
      acc[t] = __builtin_amdgcn_wmma_f32_16x16x4_f32(false, a, false, b,
                                                     (short)0, acc[t], false, false);
    }
  }
#pragma unroll
  for (int t = 0; t < 4; ++t) st_d(out, CD, m0, n0 + t * 16, lane, acc[t]);
}

// ---------------- host side ----------------
extern "C" void kernel_launch(void* const* d_in, const int* in_sizes, int n_in,
                              void* d_out, int out_size, void* d_ws, size_t ws_size,
                              hipStream_t stream) {
  const float* x    = (const float*)d_in[0];
  const float* ipw  = (const float*)d_in[1];
  const float* cw   = (const float*)d_in[2];
  const float* cb   = (const float*)d_in[3];
  const float* xpw  = (const float*)d_in[4];
  const float* dtw  = (const float*)d_in[5];
  const float* dtb  = (const float*)d_in[6];
  const float* alog = (const float*)d_in[7];
  const float* dsv  = (const float*)d_in[8];
  const float* lng  = (const float*)d_in[9];
  const float* lnb  = (const float*)d_in[10];
  const float* ow   = (const float*)d_in[11];
  float* out = (float*)d_out;

  float* wsf = (float*)d_ws;
  size_t off = 0;
  float* xssm   = wsf + off; off += (size_t)MROWS * CD;            // 6.3 MB
  float* zbuf   = wsf + off; off += (size_t)MROWS * CD;            // 6.3 MB
  float* xs     = wsf + off; off += (size_t)KD * MROWS * CD;       // 25 MB
  float* proj   = wsf + off; off += (size_t)KD * MROWS * DPROJ;    // 6.3 MB
  float* delta  = wsf + off; off += (size_t)KD * MROWS * CD;       // 25 MB (becomes cumd)
  float* ys     = wsf + off; off += (size_t)KD * MROWS * CD;       // 25 MB
  float* g      = wsf + off; off += (size_t)MROWS * CD;            // 6.3 MB
  float* chst   = wsf + off; off += (size_t)KD * BS * NCH * CD * NS; // 6.3 MB
  float* sstart = wsf + off; off += (size_t)KD * BS * NCH * CD * NS; // 6.3 MB
  (void)ws_size; (void)in_sizes; (void)n_in; (void)out_size;

  dim3 w32(32, 1, 1);
  k_inproj    <<<dim3(TWO_C / 64, MROWS / 16), w32, 0, stream>>>(x, ipw, xssm, zbuf);
  k_conv_dirs <<<(BS * LS * CD + 255) / 256, 256, 0, stream>>>(xssm, cw, cb, xs);
  k_proj      <<<dim3(MROWS / 16, KD), w32, 0, stream>>>(xs, xpw, proj);
  k_delta     <<<dim3(CD / 64, MROWS / 16, KD), w32, 0, stream>>>(proj, dtw, dtb, delta);
  k_scan_local<<<KD * BS * NCH, CD, 0, stream>>>(xs, proj, delta, alog, dsv, ys, chst);
  k_scan_carry<<<KD * BS, CD, 0, stream>>>(delta, chst, alog, sstart);
  k_scan_fix  <<<KD * BS * NCH, CD, 0, stream>>>(proj, delta, alog, sstart, ys);
  k_merge_ln  <<<MROWS, CD, 0, stream>>>(ys, zbuf, lng, lnb, g);
  k_out       <<<dim3(CD / 64, MROWS / 16), w32, 0, stream>>>(g, ow, out);
}